// ChildSumTreeLSTM_57732950393358
// MI455X (gfx1250) — compile-verified
//
#include <hip/hip_runtime.h>
#include <hip/hip_bf16.h>

// ---------------------------------------------------------------------------
// Child-Sum TreeLSTM for MI455X (gfx1250), f32 WMMA 16x16x4 path.
// Level-by-level (9 dependent launches). Leaf level skips the Wh_iou/Wfx/Wfh
// GEMMs (children are zero => exact). f32 chosen because the problem is
// bandwidth/launch-latency bound (~350MB traffic @ 23.3TB/s ~= 15us vs
// ~58 GFLOP), so exact-precision f32 WMMA costs nothing vs bf16.
// ---------------------------------------------------------------------------

typedef float v2f __attribute__((ext_vector_type(2)));
typedef float v8f __attribute__((ext_vector_type(8)));

__device__ __forceinline__ v8f wmma_f32_k4(v2f a, v2f b, v8f c) {
  // D(16x16,f32) = A(16x4,f32) * B(4x16,f32) + C
  return __builtin_amdgcn_wmma_f32_16x16x4_f32(
      /*neg_a=*/false, a, /*neg_b=*/false, b,
      /*c_mod=*/(short)0, c, /*reuse_a=*/false, /*reuse_b=*/false);
}

__device__ __forceinline__ float sigmoidf_(float v) {
  return 1.0f / (1.0f + __expf(-v));
}

// One block = 16-node M-tile, 256 threads = 8 wave32 waves.
// Each wave handles channel tiles jt = wave, wave+8 (16 channels each).
template<bool LEAF>
__global__ __launch_bounds__(256)
void tree_lstm_level(const float* __restrict__ xl,        // [n,256] level slice of x
                     const float* __restrict__ child_c,   // [4n,256] or null
                     const float* __restrict__ child_h,   // [4n,256] or null
                     const float* __restrict__ Wx_iou,    // [768,256]
                     const float* __restrict__ Wh_iou,    // [768,256]
                     const float* __restrict__ Wfx,       // [256,256]
                     const float* __restrict__ Wfh,       // [256,256]
                     const float* __restrict__ b_iou_x,   // [768]
                     const float* __restrict__ b_iou_h,   // [768]
                     const float* __restrict__ b_fx,      // [256]
                     const float* __restrict__ b_fh,      // [256]
                     float* __restrict__ out_c,           // [n,256]
                     float* __restrict__ out_h,           // [n,256]
                     int n)
{
  __shared__ float sXL[16 * 256];   // x tile
  __shared__ float sHS[16 * 256];   // child-h sum tile

  const int tid = threadIdx.x;
  const int node_base = blockIdx.x * 16;

  // Stage x tile and child-h sum tile into LDS (zero-padded past n).
  for (int idx = tid; idx < 16 * 256; idx += 256) {
    const int m = idx >> 8, ch = idx & 255;
    const int node = node_base + m;
    float xv = 0.0f, hs = 0.0f;
    if (node < n) {
      xv = xl[(size_t)node * 256 + ch];
      if (!LEAF) {
        const size_t cb = (size_t)node * 4 * 256 + ch;
        hs = child_h[cb] + child_h[cb + 256] + child_h[cb + 512] + child_h[cb + 768];
      }
    }
    sXL[idx] = xv;
    sHS[idx] = hs;
  }
  __syncthreads();

  const int lane = tid & 31;
  const int wave = tid >> 5;
  const int half = lane >> 4;   // 0: lanes 0-15, 1: lanes 16-31
  const int mrow = lane & 15;   // A-fragment row
  const int nc   = lane & 15;   // B/D-fragment column

  for (int jt = wave; jt < 16; jt += 8) {
    v8f acc_i = {}, acc_o = {}, acc_u = {}, acc_fx = {};
    v8f acc_f[4] = {{}, {}, {}, {}};
    const int wr = jt * 16 + nc;  // weight row (= output channel) for this lane

    // ---- K loop over x source: i/o/u (+ fx for internal) ----
    for (int kb = 0; kb < 256; kb += 4) {
      const int k2 = kb + half * 2;
      v2f a  = *(const v2f*)&sXL[mrow * 256 + k2];
      v2f bi = *(const v2f*)&Wx_iou[(size_t)(0   + wr) * 256 + k2];
      v2f bo = *(const v2f*)&Wx_iou[(size_t)(256 + wr) * 256 + k2];
      v2f bu = *(const v2f*)&Wx_iou[(size_t)(512 + wr) * 256 + k2];
      acc_i = wmma_f32_k4(a, bi, acc_i);
      acc_o = wmma_f32_k4(a, bo, acc_o);
      acc_u = wmma_f32_k4(a, bu, acc_u);
      if (!LEAF) {
        v2f bx = *(const v2f*)&Wfx[(size_t)wr * 256 + k2];
        acc_fx = wmma_f32_k4(a, bx, acc_fx);
      }
    }

    if (!LEAF) {
      // ---- K loop over child-h-sum source: i/o/u ----
      for (int kb = 0; kb < 256; kb += 4) {
        const int k2 = kb + half * 2;
        v2f a  = *(const v2f*)&sHS[mrow * 256 + k2];
        v2f bi = *(const v2f*)&Wh_iou[(size_t)(0   + wr) * 256 + k2];
        v2f bo = *(const v2f*)&Wh_iou[(size_t)(256 + wr) * 256 + k2];
        v2f bu = *(const v2f*)&Wh_iou[(size_t)(512 + wr) * 256 + k2];
        acc_i = wmma_f32_k4(a, bi, acc_i);
        acc_o = wmma_f32_k4(a, bo, acc_o);
        acc_u = wmma_f32_k4(a, bu, acc_u);
      }
      // ---- forget gates: per-child GEMM over child_h ----
      const int crow = min(node_base + mrow, n - 1);  // clamp; store is guarded
      #pragma unroll
      for (int k = 0; k < 4; ++k) {
        const float* hrow = child_h + (size_t)(crow * 4 + k) * 256;
        for (int kb = 0; kb < 256; kb += 4) {
          const int k2 = kb + half * 2;
          v2f a  = *(const v2f*)&hrow[k2];
          v2f bf = *(const v2f*)&Wfh[(size_t)wr * 256 + k2];
          acc_f[k] = wmma_f32_k4(a, bf, acc_f[k]);
        }
      }
    }

    // ---- elementwise gating + stores ----
    const int ch = jt * 16 + nc;
    const float bi_ = b_iou_x[ch]       + b_iou_h[ch];
    const float bo_ = b_iou_x[256 + ch] + b_iou_h[256 + ch];
    const float bu_ = b_iou_x[512 + ch] + b_iou_h[512 + ch];
    float bf_ = 0.0f;
    if (!LEAF) bf_ = b_fx[ch] + b_fh[ch];

    #pragma unroll
    for (int r = 0; r < 8; ++r) {
      const int m = r + half * 8;          // D layout: VGPR r, halves are M and M+8
      const int node = node_base + m;
      const float ig = sigmoidf_(acc_i[r] + bi_);
      const float og = sigmoidf_(acc_o[r] + bo_);
      const float ug = tanhf(acc_u[r] + bu_);
      float cc = ig * ug;
      if (!LEAF) {
        const int ndc = min(node, n - 1);
        const float fxv = acc_fx[r] + bf_;
        #pragma unroll
        for (int k = 0; k < 4; ++k) {
          const float fg = sigmoidf_(acc_f[k][r] + fxv);
          cc += fg * child_c[(size_t)(ndc * 4 + k) * 256 + ch];
        }
      }
      const float hh = og * tanhf(cc);
      if (node < n) {
        out_c[(size_t)node * 256 + ch] = cc;
        out_h[(size_t)node * 256 + ch] = hh;
      }
    }
  }
}

// ---------------------------------------------------------------------------
// Host launcher: 9 dependent level launches, ping-pong (c,h) in workspace.
// ws layout (floats): Ac[65536*256] Ah[65536*256] Bc[16384*256] Bh[16384*256]
// => 160 MB total. Root level writes directly into d_out (c | h).
// ---------------------------------------------------------------------------
extern "C" void kernel_launch(void* const* d_in, const int* in_sizes, int n_in,
                              void* d_out, int out_size, void* d_ws, size_t ws_size,
                              hipStream_t stream) {
  const float* x       = (const float*)d_in[0];
  const float* W_iou_x = (const float*)d_in[1];
  const float* b_iou_x = (const float*)d_in[2];
  const float* W_iou_h = (const float*)d_in[3];
  const float* b_iou_h = (const float*)d_in[4];
  const float* W_fx    = (const float*)d_in[5];
  const float* b_fx    = (const float*)d_in[6];
  const float* W_fh    = (const float*)d_in[7];
  const float* b_fh    = (const float*)d_in[8];
  float* out = (float*)d_out;

  float* ws = (float*)d_ws;
  const size_t szA = (size_t)65536 * 256;
  const size_t szB = (size_t)16384 * 256;
  float* Ac = ws;
  float* Ah = Ac + szA;
  float* Bc = Ah + szA;
  float* Bh = Bc + szB;

  static const int LEVEL_OFF[9] = {0, 1, 5, 21, 85, 341, 1365, 5461, 21845};

  for (int l = 8; l >= 0; --l) {
    int nl = 1;
    for (int i = 0; i < l; ++i) nl *= 4;           // 4^l nodes at this level
    const float* xl = x + (size_t)LEVEL_OFF[l] * 256;

    float *oc, *oh;
    if (l == 0)                  { oc = out; oh = out + 256; }
    else if (((8 - l) & 1) == 0) { oc = Ac;  oh = Ah; }
    else                         { oc = Bc;  oh = Bh; }

    const int blocks = (nl + 15) / 16;
    if (l == 8) {
      tree_lstm_level<true><<<blocks, 256, 0, stream>>>(
          xl, nullptr, nullptr,
          W_iou_x, W_iou_h, W_fx, W_fh,
          b_iou_x, b_iou_h, b_fx, b_fh,
          oc, oh, nl);
    } else {
      const float *cc, *chh;
      if (((8 - (l + 1)) & 1) == 0) { cc = Ac; chh = Ah; }
      else                          { cc = Bc; chh = Bh; }
      tree_lstm_level<false><<<blocks, 256, 0, stream>>>(
          xl, cc, chh,
          W_iou_x, W_iou_h, W_fx, W_fh,
          b_iou_x, b_iou_h, b_fx, b_fh,
          oc, oh, nl);
    }
  }
  (void)in_sizes; (void)n_in; (void)out_size; (void)ws_size;
}